// SinkhornDistance_81140522156488
// MI455X (gfx1250) — compile-verified
//
#include <hip/hip_runtime.h>

// Sinkhorn distance, B=4, N=2048, feat=3, eps=0.1, max_iter=100, thresh=0.1
// Outputs flat in d_out: [cost(1), pi(B*N*N), C(B*N*N)]  (float32)

constexpr int   kN    = 2048;
constexpr int   kB    = 4;
constexpr int   kBN   = kB * kN;              // 8192
constexpr size_t kBNN = (size_t)kB * kN * kN; // 16,777,216
constexpr int   kMaxIter = 100;
constexpr float kEps     = 0.1f;
constexpr float kInvEps  = 10.0f;
// err >= 0.1 continues; err = fixed/(2^24 * B). Continue iff fixed >= 0.1*4*2^24 = 6710886.4
constexpr unsigned long long kErrThreshFixed = 6710887ULL;
constexpr float  kErrScale  = 16777216.0f;     // 2^24
constexpr double kCostScale = 4294967296.0;    // 2^32

typedef float v2f __attribute__((ext_vector_type(2)));
typedef float v8f __attribute__((ext_vector_type(8)));

// ---------------- init: zero u/v, err[], cost accumulator; precompute norms ----
__global__ void init_kernel(const float* __restrict__ x, const float* __restrict__ y,
                            float* __restrict__ u, float* __restrict__ v,
                            float* __restrict__ xn, float* __restrict__ yn,
                            unsigned long long* __restrict__ err,
                            unsigned long long* __restrict__ cost_acc) {
  int p = blockIdx.x * blockDim.x + threadIdx.x;
  if (p < kBN) {
    u[p] = 0.0f; v[p] = 0.0f;
    float a0 = x[3*p], a1 = x[3*p+1], a2 = x[3*p+2];
    xn[p] = a0*a0 + a1*a1 + a2*a2;
    float b0 = y[3*p], b1 = y[3*p+1], b2 = y[3*p+2];
    yn[p] = b0*b0 + b1*b1 + b2*b2;
  }
  if (p < kMaxIter) err[p] = 0ULL;
  if (p == 0) *cost_acc = 0ULL;
}

// ---------------- C = |x|^2 + |y|^2 - 2 * X.Y^T via V_WMMA_F32_16X16X4_F32 ----
// One wave computes one 16x16 tile of C; K=4 (feature dim 3, zero padded).
__global__ void cost_wmma_kernel(const float* __restrict__ x, const float* __restrict__ y,
                                 const float* __restrict__ xn, const float* __restrict__ yn,
                                 float* __restrict__ C) {
  const int lane = threadIdx.x & 31;
  const int wave = threadIdx.x >> 5;
  const int tile = blockIdx.x * 4 + wave;   // 4 waves/block, B*128*128 tiles total
  const int bt  = tile >> 14;               // / (128*128)
  const int rem = tile & 16383;
  const int ti  = rem >> 7;                 // row tile (16 rows)
  const int tj  = rem & 127;                // col tile (16 cols)
  const int m   = lane & 15;
  const int hi  = lane >> 4;                // 0: K=0,1   1: K=2,3(pad)

  // A-matrix 16x4 f32 layout (ISA 7.12.2): lanes 0-15 -> VGPR0=K0,VGPR1=K1;
  // lanes 16-31 -> VGPR0=K2, VGPR1=K3. B-matrix (4x16 = Y^T) mirrors with N=lane&15.
  const float* xp = x + ((size_t)bt * kN + ti * 16 + m) * 3;
  const float* yp = y + ((size_t)bt * kN + tj * 16 + m) * 3;
  v2f a, b;
  if (hi == 0) { a.x = xp[0]; a.y = xp[1]; b.x = yp[0]; b.y = yp[1]; }
  else         { a.x = xp[2]; a.y = 0.0f;  b.x = yp[2]; b.y = 0.0f;  }

  v8f acc = {};
  acc = __builtin_amdgcn_wmma_f32_16x16x4_f32(false, a, false, b,
                                              (short)0, acc, false, false);

  // D layout: VGPR r holds M = r + 8*hi, N = lane&15
  const float ynv = yn[bt * kN + tj * 16 + m];
  float* cb = C + (size_t)bt * kN * kN + (size_t)(ti * 16) * kN + tj * 16;
#pragma unroll
  for (int r = 0; r < 8; ++r) {
    int row = r + 8 * hi;
    float xnr = xn[bt * kN + ti * 16 + row];
    cb[(size_t)row * kN + m] = fmaf(-2.0f, acc[r], xnr + ynv);
  }
}

// ---------------- CT[b][j][i] = C[b][i][j]  (into pi region, reused as scratch) --
__global__ void transpose_kernel(const float* __restrict__ C, float* __restrict__ CT) {
  __shared__ float tile[32][33];
  const int b  = blockIdx.z;
  const int i0 = blockIdx.y * 32;
  const int j0 = blockIdx.x * 32;
  const int tx = threadIdx.x;   // 0..31
  const int ty = threadIdx.y;   // 0..7
  const float* src = C  + (size_t)b * kN * kN;
  float*       dst = CT + (size_t)b * kN * kN;
#pragma unroll
  for (int k = 0; k < 4; ++k)
    tile[ty + k*8][tx] = src[(size_t)(i0 + ty + k*8) * kN + (j0 + tx)];
  __syncthreads();
#pragma unroll
  for (int k = 0; k < 4; ++k)
    dst[(size_t)(j0 + ty + k*8) * kN + (i0 + tx)] = tile[tx][ty + k*8];
}

// ---------------- u pass: u_i += eps*(log_mu - log(sum_j exp((u_i+v_j-C_ij)/eps)+1e-6))
__global__ void u_pass_kernel(const float* __restrict__ C, float* __restrict__ u,
                              const float* __restrict__ v,
                              unsigned long long* __restrict__ err, int t) {
  if (t > 0 && err[t - 1] < kErrThreshFixed) return;  // converged earlier
  __shared__ float red[256];
  const int row = blockIdx.x;          // b*N + i
  const int bt  = row >> 11;
  const int tid = threadIdx.x;
  const float ui = u[row];
  const float* crow = C + (size_t)row * kN;
  const float* vb   = v + (bt << 11);
  float s = 0.0f;
#pragma unroll
  for (int k = 0; k < 8; ++k) {
    int j = tid + k * 256;
    s += __expf((ui + vb[j] - crow[j]) * kInvEps);
  }
  red[tid] = s; __syncthreads();
  for (int off = 128; off > 0; off >>= 1) {
    if (tid < off) red[tid] += red[tid + off];
    __syncthreads();
  }
  if (tid == 0) {
    float lse = logf(red[0] + 1e-6f);
    float log_mu = logf(1.0f / kN + 1e-8f);
    float un = kEps * (log_mu - lse) + ui;
    u[row] = un;
    // order-independent (integer) accumulation -> deterministic convergence test
    atomicAdd(err + t, (unsigned long long)(fabsf(un - ui) * kErrScale));
  }
}

// ---------------- v pass on C^T rows (column LSE), uses freshly updated u --------
__global__ void v_pass_kernel(const float* __restrict__ CT, const float* __restrict__ u,
                              float* __restrict__ v,
                              const unsigned long long* __restrict__ err, int t) {
  if (t > 0 && err[t - 1] < kErrThreshFixed) return;
  __shared__ float red[256];
  const int row = blockIdx.x;          // b*N + j
  const int bt  = row >> 11;
  const int tid = threadIdx.x;
  const float vj = v[row];
  const float* ctrow = CT + (size_t)row * kN;
  const float* ub    = u + (bt << 11);
  float s = 0.0f;
#pragma unroll
  for (int k = 0; k < 8; ++k) {
    int i = tid + k * 256;
    s += __expf((vj + ub[i] - ctrow[i]) * kInvEps);
  }
  red[tid] = s; __syncthreads();
  for (int off = 128; off > 0; off >>= 1) {
    if (tid < off) red[tid] += red[tid + off];
    __syncthreads();
  }
  if (tid == 0) {
    float lse = logf(red[0] + 1e-6f);
    float log_nu = logf(1.0f / kN + 1e-8f);
    v[row] = kEps * (log_nu - lse) + vj;
  }
}

// ---------------- pi = exp(M), cost += pi*C (fixed-point, deterministic) ---------
__global__ void pi_kernel(const float* __restrict__ C, const float* __restrict__ u,
                          const float* __restrict__ v, float* __restrict__ pi,
                          unsigned long long* __restrict__ cost_acc) {
  __shared__ float red[256];
  const int tid = threadIdx.x;
  const size_t base = (size_t)blockIdx.x * 2048;
  float local = 0.0f;
#pragma unroll
  for (int k = 0; k < 8; ++k) {
    size_t idx = base + tid + (size_t)k * 256;
    float cv = C[idx];
    int j  = (int)(idx & 2047);
    int r  = (int)(idx >> 11);        // b*N + i
    int bt = r >> 11;
    float p = __expf((u[r] + v[(bt << 11) + j] - cv) * kInvEps);
    pi[idx] = p;
    local += p * cv;
  }
  red[tid] = local; __syncthreads();
  for (int off = 128; off > 0; off >>= 1) {
    if (tid < off) red[tid] += red[tid + off];
    __syncthreads();
  }
  if (tid == 0)
    atomicAdd(cost_acc, (unsigned long long)((double)red[0] * kCostScale));
}

__global__ void finalize_kernel(const unsigned long long* __restrict__ cost_acc,
                                float* __restrict__ out) {
  out[0] = (float)((double)(*cost_acc) / (kCostScale * (double)kB));
}

// =================================================================================
extern "C" void kernel_launch(void* const* d_in, const int* in_sizes, int n_in,
                              void* d_out, int out_size, void* d_ws, size_t ws_size,
                              hipStream_t stream) {
  (void)in_sizes; (void)n_in; (void)out_size; (void)ws_size;
  const float* x = (const float*)d_in[0];
  const float* y = (const float*)d_in[1];

  float* out = (float*)d_out;
  float* pi  = out + 1;
  float* C   = out + 1 + kBNN;

  float* u  = (float*)d_ws;
  float* v  = u + kBN;
  float* xn = v + kBN;
  float* yn = xn + kBN;
  unsigned long long* err      = (unsigned long long*)(yn + kBN); // 8B aligned
  unsigned long long* cost_acc = err + kMaxIter;

  float* CT = pi;  // pi output region doubles as C^T scratch during iterations

  init_kernel<<<(kBN + 255) / 256, 256, 0, stream>>>(x, y, u, v, xn, yn, err, cost_acc);
  cost_wmma_kernel<<<kB * 128 * 128 / 4, 128, 0, stream>>>(x, y, xn, yn, C);
  transpose_kernel<<<dim3(64, 64, kB), dim3(32, 8), 0, stream>>>(C, CT);

  for (int t = 0; t < kMaxIter; ++t) {
    u_pass_kernel<<<kBN, 256, 0, stream>>>(C, u, v, err, t);
    v_pass_kernel<<<kBN, 256, 0, stream>>>(CT, u, v, err, t);
  }

  pi_kernel<<<(int)(kBNN / 2048), 256, 0, stream>>>(C, u, v, pi, cost_acc);
  finalize_kernel<<<1, 1, 0, stream>>>(cost_acc, out);
}